// myDGI_81819126988985
// MI455X (gfx1250) — compile-verified
//
#include <hip/hip_runtime.h>
#include <cstddef>

#define N_USERS 100000
#define N_ITEMS 100000
#define DD      128
#define DEG     16
#define BATCH   100000
#define ALPHA   0.1f
#define BM      32

typedef __attribute__((ext_vector_type(2))) float v2f;
typedef __attribute__((ext_vector_type(8))) float v8f;

__device__ __forceinline__ float sigf(float x) { return 1.0f / (1.0f + expf(-x)); }

// ---------------------------------------------------------------------------
// Column-mean partials: each block sums 256 rows for all 128 columns.
// ---------------------------------------------------------------------------
__global__ __launch_bounds__(128) void colmean_partial_k(
    const float* __restrict__ feat, int nrows, float* __restrict__ partial) {
  const int c  = threadIdx.x;
  const int r0 = blockIdx.x * 256;
  int r1 = r0 + 256; if (r1 > nrows) r1 = nrows;
  float acc = 0.f;
  for (int r = r0; r < r1; ++r) acc += feat[(size_t)r * DD + c];
  partial[(size_t)blockIdx.x * DD + c] = acc;
}

// Deterministic fixed-order reduction of [nparts][128] partials.
__global__ __launch_bounds__(128) void reduce_partial_k(
    const float* __restrict__ partial, int nparts, float scale, float* __restrict__ out) {
  const int c = threadIdx.x;
  float acc = 0.f;
  for (int p = 0; p < nparts; ++p) acc += partial[(size_t)p * DD + c];
  out[c] = acc * scale;
}

// ---------------------------------------------------------------------------
// Fused graph pass over one bipartite adjacency (fixed degree 16, sorted src).
// One wave per node: computes
//   - rating-weighted neighbor mean (extract numerator / den)  -> block partial
//   - unweighted neighbor mean of real features (GAT low)      -> low_r
//   - unweighted neighbor mean of fake features                -> low_f
//   - sigmoid(leakyrelu([self, low] . att_w + b)) gates        -> gate_r/gate_f
// Neighbor rows are read ONCE and serve both extract and GAT.
// ---------------------------------------------------------------------------
__global__ __launch_bounds__(256) void graph_pass_k(
    const float* __restrict__ nbr,   const float* __restrict__ nbf,
    const float* __restrict__ selfr, const float* __restrict__ selff,
    const int*   __restrict__ dst,   const float* __restrict__ val,
    const float* __restrict__ att_w, const float* __restrict__ att_b,
    float* __restrict__ low_r,  float* __restrict__ low_f,
    float* __restrict__ gate_r, float* __restrict__ gate_f,
    float* __restrict__ partial) {
  __shared__ float sl[8][DD + 4];
  const int lane = threadIdx.x & 31;
  const int wv   = threadIdx.x >> 5;
  const int node = blockIdx.x * 8 + wv;
  const size_t ebase = (size_t)node * DEG;
  const int co = 4 * lane;

  int   jv  = 0;
  float wvv = 0.f;
  if (lane < DEG) { jv = dst[ebase + lane]; wvv = val[ebase + lane]; }

  float4 aext = {0, 0, 0, 0}, alow = {0, 0, 0, 0}, alowf = {0, 0, 0, 0};
  float den = 0.f;
  for (int e = 0; e < DEG; ++e) {
    const int   j = __shfl(jv, e, 32);
    const float w = __shfl(wvv, e, 32);
    const float4 f  = *reinterpret_cast<const float4*>(nbr + (size_t)j * DD + co);
    const float4 ff = *reinterpret_cast<const float4*>(nbf + (size_t)j * DD + co);
    aext.x += w * f.x; aext.y += w * f.y; aext.z += w * f.z; aext.w += w * f.w;
    alow.x += f.x;  alow.y += f.y;  alow.z += f.z;  alow.w += f.w;
    alowf.x += ff.x; alowf.y += ff.y; alowf.z += ff.z; alowf.w += ff.w;
    den += w;
  }

  const float inv16 = 1.0f / 16.0f;
  const float4 low  = {alow.x * inv16,  alow.y * inv16,  alow.z * inv16,  alow.w * inv16};
  const float4 lowf = {alowf.x * inv16, alowf.y * inv16, alowf.z * inv16, alowf.w * inv16};
  *reinterpret_cast<float4*>(low_r + (size_t)node * DD + co) = low;
  *reinterpret_cast<float4*>(low_f + (size_t)node * DD + co) = lowf;

  const float4 aw0 = *reinterpret_cast<const float4*>(att_w + co);
  const float4 aw1 = *reinterpret_cast<const float4*>(att_w + DD + co);
  const float  ab  = att_b[0];

  // real gate
  {
    const float4 sf = *reinterpret_cast<const float4*>(selfr + (size_t)node * DD + co);
    float sp = aw0.x * sf.x + aw0.y * sf.y + aw0.z * sf.z + aw0.w * sf.w
             + aw1.x * low.x + aw1.y * low.y + aw1.z * low.z + aw1.w * low.w;
    for (int o = 16; o >= 1; o >>= 1) sp += __shfl_down(sp, o, 32);
    if (lane == 0) {
      float s = sp + ab;
      s = (s >= 0.f) ? s : ALPHA * s;
      gate_r[node] = sigf(s);
    }
  }
  // fake gate
  {
    const float4 sf = *reinterpret_cast<const float4*>(selff + (size_t)node * DD + co);
    float sp = aw0.x * sf.x + aw0.y * sf.y + aw0.z * sf.z + aw0.w * sf.w
             + aw1.x * lowf.x + aw1.y * lowf.y + aw1.z * lowf.z + aw1.w * lowf.w;
    for (int o = 16; o >= 1; o >>= 1) sp += __shfl_down(sp, o, 32);
    if (lane == 0) {
      float s = sp + ab;
      s = (s >= 0.f) ? s : ALPHA * s;
      gate_f[node] = sigf(s);
    }
  }

  // extract: per-node normalized weighted mean -> per-block partial (deterministic)
  const float dd  = (den > 0.f) ? den : 1.0f;
  const float inv = 1.0f / dd;
  sl[wv][co + 0] = aext.x * inv;
  sl[wv][co + 1] = aext.y * inv;
  sl[wv][co + 2] = aext.z * inv;
  sl[wv][co + 3] = aext.w * inv;
  __syncthreads();
  if (threadIdx.x < DD) {
    float a = 0.f;
    for (int w8 = 0; w8 < 8; ++w8) a += sl[w8][threadIdx.x];
    partial[(size_t)blockIdx.x * DD + threadIdx.x] = a;
  }
}

// ---------------------------------------------------------------------------
// Tiny head: g/h/S_two/S and v = disc_w @ S. Single block, 128 threads.
// ---------------------------------------------------------------------------
__global__ __launch_bounds__(128) void head_k(
    const float* __restrict__ Su, const float* __restrict__ Si,
    const float* __restrict__ gi, const float* __restrict__ gu,
    const float* __restrict__ l1w, const float* __restrict__ l1b,
    const float* __restrict__ l2w, const float* __restrict__ l2b,
    const float* __restrict__ lw,  const float* __restrict__ lb,
    const float* __restrict__ dw,  float* __restrict__ vout) {
  __shared__ float t[DD], S[DD];
  const int d = threadIdx.x;
  float g = l1b[d], h = l2b[d];
  const float* r1 = l1w + (size_t)d * (2 * DD);
  const float* r2 = l2w + (size_t)d * (2 * DD);
  for (int k = 0; k < DD; ++k) {
    g += r1[k] * Su[k] + r1[DD + k] * gi[k];
    h += r2[k] * Si[k] + r2[DD + k] * gu[k];
  }
  t[d] = 0.5f * (g + h);
  __syncthreads();
  float s = lb[d];
  const float* rw = lw + (size_t)d * DD;
  for (int k = 0; k < DD; ++k) s += rw[k] * t[k];
  S[d] = sigf(s);
  __syncthreads();
  float vv = 0.f;
  const float* rd = dw + (size_t)d * DD;
  for (int e = 0; e < DD; ++e) vv += rd[e] * S[e];
  vout[d] = vv;
}

// ---------------------------------------------------------------------------
// Batch kernel: gather + gate-blend x=[ru,ri] (256), y=sigmoid(x@W_sub^T+b),
// prob=sigmoid(y.v + disc_b). GEMM via V_WMMA_F32_16X16X4_F32.
// Block: 256 threads (8 waves), 32 batch rows; wave w owns N-tile [16w,16w+16)
// and two 16x16 f32 accumulators (rows 0-15 and 16-31).
// ---------------------------------------------------------------------------
__global__ __launch_bounds__(256) void batch_prob_k(
    const float* __restrict__ uself, const float* __restrict__ iself,
    const float* __restrict__ low_u, const float* __restrict__ low_i,
    const float* __restrict__ gate_u, const float* __restrict__ gate_i,
    const int*   __restrict__ uidx,  const int* __restrict__ iidx,
    const float* __restrict__ wsub,  const float* __restrict__ bsub,
    const float* __restrict__ v,     const float* __restrict__ discb,
    float* __restrict__ prob) {
  __shared__ float X[BM][2 * DD + 4];   // padded stride 260: conflict-free A reads
  __shared__ float Wl[DD][36];          // padded stride 36: conflict-free B reads
  __shared__ float Pp[BM][8];
  __shared__ float vb[DD], bb[DD];

  const int tid  = threadIdx.x;
  const int lane = tid & 31;
  const int wv   = tid >> 5;
  const int bbase = blockIdx.x * BM;

  // ---- stage X = gate*self + (1-gate)*low for 32 rows, 256 cols ----
  {
    const int j = tid >> 3, seg = tid & 7;
    const int b = bbase + j;
    if (seg < 4) {
      const int u = uidx[b];
      const float g = gate_u[u], om = 1.f - g;
      const float* a = uself + (size_t)u * DD;
      const float* l = low_u + (size_t)u * DD;
      const int c0 = seg * 32;
      for (int i = 0; i < 32; i += 4) {
        const float4 fa = *reinterpret_cast<const float4*>(a + c0 + i);
        const float4 fl = *reinterpret_cast<const float4*>(l + c0 + i);
        X[j][c0 + i + 0] = g * fa.x + om * fl.x;
        X[j][c0 + i + 1] = g * fa.y + om * fl.y;
        X[j][c0 + i + 2] = g * fa.z + om * fl.z;
        X[j][c0 + i + 3] = g * fa.w + om * fl.w;
      }
    } else {
      const int it = iidx[b];
      const float g = gate_i[it], om = 1.f - g;
      const float* a = iself + (size_t)it * DD;
      const float* l = low_i + (size_t)it * DD;
      const int c0 = (seg - 4) * 32;
      for (int i = 0; i < 32; i += 4) {
        const float4 fa = *reinterpret_cast<const float4*>(a + c0 + i);
        const float4 fl = *reinterpret_cast<const float4*>(l + c0 + i);
        X[j][DD + c0 + i + 0] = g * fa.x + om * fl.x;
        X[j][DD + c0 + i + 1] = g * fa.y + om * fl.y;
        X[j][DD + c0 + i + 2] = g * fa.z + om * fl.z;
        X[j][DD + c0 + i + 3] = g * fa.w + om * fl.w;
      }
    }
  }
  if (tid < DD) { vb[tid] = v[tid]; bb[tid] = bsub[tid]; }
  __syncthreads();

  v8f c0acc = {0, 0, 0, 0, 0, 0, 0, 0};
  v8f c1acc = {0, 0, 0, 0, 0, 0, 0, 0};
  const int nbase = wv * 16;
  const int msel  = lane & 15;
  const int kof   = (lane >> 4) * 2;   // lanes 0-15: K {0,1}; lanes 16-31: K {2,3}

  for (int kc = 0; kc < 2 * DD; kc += 32) {
    // stage W tile: Wl[d][kk] = W_sub[d][kc+kk]  (B matrix is W^T)
    {
      const int d = tid >> 1, hf = tid & 1;
      const float* src = wsub + (size_t)d * (2 * DD) + kc + hf * 16;
      float* dl = &Wl[d][hf * 16];
      for (int i = 0; i < 16; i += 4) {
        const float4 f = *reinterpret_cast<const float4*>(src + i);
        dl[i + 0] = f.x; dl[i + 1] = f.y; dl[i + 2] = f.z; dl[i + 3] = f.w;
      }
    }
    __syncthreads();
    for (int k4 = 0; k4 < 32; k4 += 4) {
      v2f a0, a1, bfrag;
      const int kg = kc + k4 + kof;
      a0.x = X[msel][kg];          a0.y = X[msel][kg + 1];
      a1.x = X[16 + msel][kg];     a1.y = X[16 + msel][kg + 1];
      bfrag.x = Wl[nbase + msel][k4 + kof];
      bfrag.y = Wl[nbase + msel][k4 + kof + 1];
      c0acc = __builtin_amdgcn_wmma_f32_16x16x4_f32(false, a0, false, bfrag,
                                                    (short)0, c0acc, false, false);
      c1acc = __builtin_amdgcn_wmma_f32_16x16x4_f32(false, a1, false, bfrag,
                                                    (short)0, c1acc, false, false);
    }
    __syncthreads();
  }

  // ---- fused epilogue: sigmoid(y+b)*v[n], reduce over this wave's 16 cols ----
  const float vn = vb[nbase + msel];
  const float bn = bb[nbase + msel];
  const int rowoff = (lane >> 4) * 8;   // C layout: vgpr r = row r (+8 for hi lanes)
  for (int r = 0; r < 8; ++r) {
    float y0 = sigf(c0acc[r] + bn) * vn;
    float y1 = sigf(c1acc[r] + bn) * vn;
    for (int m = 1; m < 16; m <<= 1) {
      y0 += __shfl_xor(y0, m, 32);
      y1 += __shfl_xor(y1, m, 32);
    }
    if (msel == 0) {
      Pp[r + rowoff][wv]      = y0;
      Pp[16 + r + rowoff][wv] = y1;
    }
  }
  __syncthreads();
  if (tid < BM) {
    float tot = discb[0];
    for (int w8 = 0; w8 < 8; ++w8) tot += Pp[tid][w8];   // fixed order across N-tiles
    prob[bbase + tid] = sigf(tot);
  }
}

__global__ __launch_bounds__(256) void label_k(float* __restrict__ out) {
  const int i = blockIdx.x * 256 + threadIdx.x;
  if (i < 2 * BATCH) out[2 * BATCH + i] = (i < BATCH) ? 1.0f : 0.0f;
}

// ---------------------------------------------------------------------------
extern "C" void kernel_launch(void* const* d_in, const int* in_sizes, int n_in,
                              void* d_out, int out_size, void* d_ws, size_t ws_size,
                              hipStream_t stream) {
  (void)in_sizes; (void)n_in; (void)out_size; (void)ws_size;
  const float* uh   = (const float*)d_in[0];
  const float* ih   = (const float*)d_in[1];
  const float* fuh  = (const float*)d_in[2];
  const float* fih  = (const float*)d_in[3];
  const int*   uOne = (const int*)d_in[4];
  const int*   iOne = (const int*)d_in[5];
  const int*   uv_dst = (const int*)d_in[7];
  const float* uv_val = (const float*)d_in[8];
  const int*   vu_dst = (const int*)d_in[10];
  const float* vu_val = (const float*)d_in[11];
  const float* l1w = (const float*)d_in[12]; const float* l1b = (const float*)d_in[13];
  const float* l2w = (const float*)d_in[14]; const float* l2b = (const float*)d_in[15];
  const float* lw  = (const float*)d_in[16]; const float* lb  = (const float*)d_in[17];
  const float* lsw = (const float*)d_in[18]; const float* lsb = (const float*)d_in[19];
  const float* auw = (const float*)d_in[20]; const float* aub = (const float*)d_in[21];
  const float* aiw = (const float*)d_in[22]; const float* aib = (const float*)d_in[23];
  const float* dw  = (const float*)d_in[24]; const float* db  = (const float*)d_in[25];
  float* out = (float*)d_out;

  // workspace layout (floats)
  float* ws = (float*)d_ws;
  float* Su     = ws;            // 128
  float* Si     = ws + 128;      // 128
  float* g_item = ws + 256;      // 128
  float* g_user = ws + 384;      // 128
  float* vvec   = ws + 512;      // 128
  float* gate_u_real = ws + 1024;
  float* gate_u_fake = gate_u_real + N_USERS;
  float* gate_i_real = gate_u_fake + N_USERS;
  float* gate_i_fake = gate_i_real + N_ITEMS;
  float* low_u_real  = gate_i_fake + N_ITEMS;
  float* low_u_fake  = low_u_real + (size_t)N_USERS * DD;
  float* low_i_real  = low_u_fake + (size_t)N_USERS * DD;
  float* low_i_fake  = low_i_real + (size_t)N_ITEMS * DD;
  float* partial     = low_i_fake + (size_t)N_ITEMS * DD;  // max 12500*128

  // S_u, S_i (deterministic two-stage column means)
  const int cm_parts = (N_USERS + 255) / 256;
  colmean_partial_k<<<cm_parts, 128, 0, stream>>>(uh, N_USERS, partial);
  reduce_partial_k<<<1, 128, 0, stream>>>(partial, cm_parts, 1.0f / N_USERS, Su);
  colmean_partial_k<<<cm_parts, 128, 0, stream>>>(ih, N_ITEMS, partial);
  reduce_partial_k<<<1, 128, 0, stream>>>(partial, cm_parts, 1.0f / N_ITEMS, Si);

  // fused uv pass: extract(item) + real/fake user GAT lows + gates
  graph_pass_k<<<N_USERS / 8, 256, 0, stream>>>(
      ih, fih, uh, fuh, uv_dst, uv_val, auw, aub,
      low_u_real, low_u_fake, gate_u_real, gate_u_fake, partial);
  reduce_partial_k<<<1, 128, 0, stream>>>(partial, N_USERS / 8, 1.0f / N_USERS, g_item);

  // fused vu pass
  graph_pass_k<<<N_ITEMS / 8, 256, 0, stream>>>(
      uh, fuh, ih, fih, vu_dst, vu_val, aiw, aib,
      low_i_real, low_i_fake, gate_i_real, gate_i_fake, partial);
  reduce_partial_k<<<1, 128, 0, stream>>>(partial, N_ITEMS / 8, 1.0f / N_ITEMS, g_user);

  // head: S_two -> S -> v = disc_w @ S
  head_k<<<1, 128, 0, stream>>>(Su, Si, g_item, g_user, l1w, l1b, l2w, l2b,
                                lw, lb, dw, vvec);

  // probs (real then fake) + labels
  batch_prob_k<<<BATCH / BM, 256, 0, stream>>>(
      uh, ih, low_u_real, low_i_real, gate_u_real, gate_i_real,
      uOne, iOne, lsw, lsb, vvec, db, out);
  batch_prob_k<<<BATCH / BM, 256, 0, stream>>>(
      fuh, fih, low_u_fake, low_i_fake, gate_u_fake, gate_i_fake,
      uOne, iOne, lsw, lsb, vvec, db, out + BATCH);
  label_k<<<(2 * BATCH + 255) / 256, 256, 0, stream>>>(out);
}